// NodeConv_35880156791043
// MI455X (gfx1250) — compile-verified
//
#include <hip/hip_runtime.h>

// Problem constants (from reference): B=4, N=256, NIN=NOUT=128
#define NB   4
#define NN   256
#define NINC 128
#define NOUTC 128

typedef __attribute__((ext_vector_type(2))) float v2f;
typedef __attribute__((ext_vector_type(4))) float v4f;
typedef __attribute__((ext_vector_type(8))) float v8f;

// ---------------------------------------------------------------------------
// Kernel 1: vi = Wi @ node[b], vj = Wj @ node[b]   (f32 WMMA 16x16x4)
// One wave (32 threads) per 16x16 output tile; both weight matrices share the
// same B-fragment (node tile) loads.
//   A (16x4 f32): lanes 0-15 -> M=lane, regs {K=0,K=1}; lanes 16-31 -> {K=2,K=3}
//   B (4x16 f32): lanes 0-15 -> N=lane, regs {K=0,K=1}; lanes 16-31 -> {K=2,K=3}
//   C/D 16x16 f32: reg r, lanes 0-15 -> M=r, lanes 16-31 -> M=r+8; N=lane&15
// ---------------------------------------------------------------------------
__global__ __launch_bounds__(32) void nodeconv_gemm(
    const float* __restrict__ node,   // [NB][NINC][NN]
    const float* __restrict__ Wi,     // [NOUTC][NINC]
    const float* __restrict__ Wj,     // [NOUTC][NINC]
    float* __restrict__ vi,           // [NB][NOUTC][NN]
    float* __restrict__ vj)           // [NB][NOUTC][NN]
{
    const int tile = blockIdx.x;          // 0 .. NB*8*16-1
    const int b    = tile >> 7;
    const int t    = tile & 127;
    const int m0   = (t >> 4) << 4;       // output-channel tile base (0..112)
    const int n0   = (t & 15) << 4;       // column tile base (0..240)

    const int lane = threadIdx.x;         // 0..31 (wave32, EXEC all ones)
    const int half = lane >> 4;           // 0 or 1
    const int idx  = lane & 15;

    const float* nb = node + b * (NINC * NN);

    v8f acc_i = {0.f, 0.f, 0.f, 0.f, 0.f, 0.f, 0.f, 0.f};
    v8f acc_j = {0.f, 0.f, 0.f, 0.f, 0.f, 0.f, 0.f, 0.f};

    const int arow = (m0 + idx) * NINC;   // A fragment row (row-major weights)

    for (int kk = 0; kk < NINC; kk += 4) {
        const int ka = kk + 2 * half;     // this lane-half's K base
        // A fragments: 2 consecutive f32 -> one b64 load each
        v2f a_i = *(const v2f*)(Wi + arow + ka);
        v2f a_j = *(const v2f*)(Wj + arow + ka);
        // B fragment: column n0+idx of node, rows ka, ka+1
        v2f bf;
        bf.x = nb[(ka    ) * NN + n0 + idx];
        bf.y = nb[(ka + 1) * NN + n0 + idx];

        acc_i = __builtin_amdgcn_wmma_f32_16x16x4_f32(
            /*neg_a=*/false, a_i, /*neg_b=*/false, bf,
            /*c_mod=*/(short)0, acc_i, /*reuse_a=*/false, /*reuse_b=*/false);
        acc_j = __builtin_amdgcn_wmma_f32_16x16x4_f32(
            /*neg_a=*/false, a_j, /*neg_b=*/false, bf,
            /*c_mod=*/(short)0, acc_j, /*reuse_a=*/false, /*reuse_b=*/false);
    }

    #pragma unroll
    for (int r = 0; r < 8; ++r) {
        const int o = m0 + 8 * half + r;
        vi[(b * NOUTC + o) * NN + n0 + idx] = acc_i[r];
        vj[(b * NOUTC + o) * NN + n0 + idx] = acc_j[r];
    }
}

// ---------------------------------------------------------------------------
// Kernel 2: out[b,o,i,j] = adj[b,i,j] * (i==j ? vi[b,o,j] : vj[b,o,j])
// Pure store-bandwidth kernel: 128 MB write-once output stream in float4s,
// stored non-temporally so it does not evict the L2-resident adj/vi/vj
// (which are reused 128x/256x across blocks).
// 4 rows per 256-thread block; 64 threads per row; one float4 per thread.
// ---------------------------------------------------------------------------
__global__ __launch_bounds__(256) void nodeconv_bcast(
    const float* __restrict__ adj,    // [NB][NN][NN]
    const float* __restrict__ vi,     // [NB][NOUTC][NN]
    const float* __restrict__ vj,     // [NB][NOUTC][NN]
    float* __restrict__ out)          // [NB][NOUTC][NN][NN]
{
    const int tid    = threadIdx.x;
    const int rlocal = tid >> 6;              // 0..3  (row within block)
    const int jv     = (tid & 63) << 2;       // float4 column base 0..252

    const int row = blockIdx.x * 4 + rlocal;  // 0 .. NB*NOUTC*NN-1
    const int b   = row >> 15;                // / (NOUTC*NN)
    const int rem = row & 32767;
    const int o   = rem >> 8;
    const int i   = rem & 255;

    const v4f a4 = *(const v4f*)(adj + (b * NN + i) * NN + jv);
    const v4f w4 = *(const v4f*)(vj + (b * NOUTC + o) * NN + jv);

    v4f r4 = a4 * w4;

    // Diagonal element of this row falls in our chunk -> use vi instead of vj
    if (i >= jv && i < jv + 4) {
        const float dvi = vi[(b * NOUTC + o) * NN + i];
        r4[i - jv] = a4[i - jv] * dvi;
    }

    v4f* dst = (v4f*)(out + (((long long)(b * NOUTC + o) * NN + i) * NN) + jv);
    __builtin_nontemporal_store(r4, dst);   // streaming store (TH=NT)
}

extern "C" void kernel_launch(void* const* d_in, const int* in_sizes, int n_in,
                              void* d_out, int out_size, void* d_ws, size_t ws_size,
                              hipStream_t stream) {
    (void)in_sizes; (void)n_in; (void)out_size; (void)ws_size;

    const float* adj  = (const float*)d_in[0];  // [4,1,256,256]
    const float* node = (const float*)d_in[1];  // [4,128,256]
    const float* Wi   = (const float*)d_in[2];  // [128,128]
    const float* Wj   = (const float*)d_in[3];  // [128,128]
    float* out = (float*)d_out;                 // [4,128,256,256]

    float* vi = (float*)d_ws;                       // NB*NOUTC*NN floats
    float* vj = vi + (size_t)NB * NOUTC * NN;       // NB*NOUTC*NN floats

    // Kernel 1: 512 tiles (4 batches x 8 M-tiles x 16 N-tiles), 1 wave each
    nodeconv_gemm<<<NB * 8 * 16, 32, 0, stream>>>(node, Wi, Wj, vi, vj);

    // Kernel 2: NB*NOUTC*NN rows / 4 rows per block
    const int rows   = NB * NOUTC * NN;             // 131072
    const int blocks = rows / 4;                    // 32768
    nodeconv_bcast<<<blocks, 256, 0, stream>>>(adj, vi, vj, out);
}